// GraphSAGE_85547158602127
// MI455X (gfx1250) — compile-verified
//
#include <hip/hip_runtime.h>

#define NNODES 100000
#define NEDGES 1600000
#define NFEAT  64
#define NHID   64

typedef float v2f __attribute__((ext_vector_type(2)));
typedef float v8f __attribute__((ext_vector_type(8)));

// ---------------------------------------------------------------- zero fill
__global__ __launch_bounds__(256) void zero_f32(float* __restrict__ p, int n) {
    int i = blockIdx.x * blockDim.x + threadIdx.x;
    int stride = gridDim.x * blockDim.x;
    for (; i < n; i += stride) p[i] = 0.0f;
}

// ------------------------------------------------- edge gather + atomic scatter
// One wave (32 lanes) per edge. Lane l moves feature columns 2l, 2l+1.
// Optionally accumulates float degree (lane 0).
__global__ __launch_bounds__(256) void edge_scatter(
    const float* __restrict__ feat,            // [N, 64]
    const long long* __restrict__ src,
    const long long* __restrict__ dst,
    float* __restrict__ agg,                   // [N, 64] accumulators
    float* __restrict__ deg,                   // [N] or nullptr
    int nedges)
{
    int edge = blockIdx.x * (blockDim.x >> 5) + (threadIdx.x >> 5);
    if (edge >= nedges) return;                // uniform per wave
    int lane = threadIdx.x & 31;

    long long s = src[edge];
    long long d = dst[edge];

    const float2* fp = reinterpret_cast<const float2*>(feat + s * 64);
    float2 v = fp[lane];

    float* outp = agg + d * 64 + lane * 2;
    atomicAdd(outp,     v.x);
    atomicAdd(outp + 1, v.y);

    if (deg != nullptr && lane == 0) atomicAdd(deg + d, 1.0f);
}

// ------------------------------------------------- layer 1: fused WMMA SAGEConv
// h = relu( (agg * invdeg) @ W_l + b + x @ W_r ),  all fp32 via WMMA 16x16x4.
// One wave per 16-node tile; 4 accumulators cover all 64 output columns.
__global__ __launch_bounds__(256) void sage_layer1_wmma(
    const float* __restrict__ agg,   // [N,64] neighbor sums
    const float* __restrict__ x,     // [N,64]
    const float* __restrict__ deg,   // [N]
    const float* __restrict__ W_l,   // [64,64] row-major (k, n)
    const float* __restrict__ bvec,  // [64]
    const float* __restrict__ W_r,   // [64,64]
    float* __restrict__ hout,        // [N,64]
    int ntiles)
{
    int tile = blockIdx.x * (blockDim.x >> 5) + (threadIdx.x >> 5);
    if (tile >= ntiles) return;      // uniform per wave -> EXEC all-1s for WMMA
    int lane = threadIdx.x & 31;
    int m    = lane & 15;            // A row / B,C column within tile
    int half = lane >> 4;            // 0: K={k,k+1}   1: K={k+2,k+3}

    int row0 = tile * 16;
    float invd = 1.0f / fmaxf(deg[row0 + m], 1.0f);

    const float* aggRow = agg + (long long)(row0 + m) * 64 + 2 * half;
    const float* xRow   = x   + (long long)(row0 + m) * 64 + 2 * half;

    v8f c0 = {}, c1 = {}, c2 = {}, c3 = {};

    #pragma unroll
    for (int k0 = 0; k0 < 64; k0 += 4) {
        // A fragments (16x4 fp32): lane holds A[m][k0+2*half .. +1]
        v2f aAgg, aX;
        aAgg.x = aggRow[k0]     * invd;
        aAgg.y = aggRow[k0 + 1] * invd;
        aX.x   = xRow[k0];
        aX.y   = xRow[k0 + 1];

        int krow = k0 + 2 * half;
        const float* wl0 = W_l + krow * 64;       // row k
        const float* wl1 = wl0 + 64;              // row k+1
        const float* wr0 = W_r + krow * 64;
        const float* wr1 = wr0 + 64;

        // B fragments (4x16 fp32): lane holds B[krow][n0+m], B[krow+1][n0+m]
        #define DO_NTILE(CACC, N0)                                              \
        {                                                                       \
            v2f bl, br;                                                         \
            bl.x = wl0[(N0) + m]; bl.y = wl1[(N0) + m];                         \
            br.x = wr0[(N0) + m]; br.y = wr1[(N0) + m];                         \
            CACC = __builtin_amdgcn_wmma_f32_16x16x4_f32(                       \
                false, aAgg, false, bl, (short)0, CACC, false, false);          \
            CACC = __builtin_amdgcn_wmma_f32_16x16x4_f32(                       \
                false, aX,   false, br, (short)0, CACC, false, false);          \
        }
        DO_NTILE(c0, 0)
        DO_NTILE(c1, 16)
        DO_NTILE(c2, 32)
        DO_NTILE(c3, 48)
        #undef DO_NTILE
    }

    // Epilogue: bias + ReLU + store.
    // C/D layout: VGPR r -> row (r + 8*half), column (n0 + m).
    v8f cs[4] = { c0, c1, c2, c3 };
    #pragma unroll
    for (int nt = 0; nt < 4; nt++) {
        int n = nt * 16 + m;
        float bias = bvec[n];
        #pragma unroll
        for (int r = 0; r < 8; r++) {
            int row = row0 + r + 8 * half;
            float v = cs[nt][r] + bias;
            v = fmaxf(v, 0.0f);
            hout[(long long)row * 64 + n] = v;
        }
    }
}

// ------------------------------------------------- layer 2: fused matvec output
// out[i] = dot(agg2[i]*invdeg, W2_l) + b2 + dot(h[i], W2_r)
__global__ __launch_bounds__(256) void sage_out(
    const float* __restrict__ agg2,  // [N,64] neighbor sums of h
    const float* __restrict__ h,     // [N,64]
    const float* __restrict__ deg,   // [N]
    const float* __restrict__ W2_l,  // [64]
    const float* __restrict__ b2,    // [1]
    const float* __restrict__ W2_r,  // [64]
    float* __restrict__ out, int n)
{
    int i = blockIdx.x * blockDim.x + threadIdx.x;
    if (i >= n) return;
    float invd = 1.0f / fmaxf(deg[i], 1.0f);
    const float4* a4 = reinterpret_cast<const float4*>(agg2 + (long long)i * 64);
    const float4* h4 = reinterpret_cast<const float4*>(h    + (long long)i * 64);
    const float4* wl = reinterpret_cast<const float4*>(W2_l);
    const float4* wr = reinterpret_cast<const float4*>(W2_r);
    float accA = 0.0f, accH = 0.0f;
    #pragma unroll
    for (int k = 0; k < 16; k++) {
        float4 av = a4[k], hv = h4[k], l = wl[k], r = wr[k];
        accA += av.x * l.x + av.y * l.y + av.z * l.z + av.w * l.w;
        accH += hv.x * r.x + hv.y * r.y + hv.z * r.z + hv.w * r.w;
    }
    out[i] = accA * invd + accH + b2[0];
}

// ---------------------------------------------------------------- launcher
extern "C" void kernel_launch(void* const* d_in, const int* in_sizes, int n_in,
                              void* d_out, int out_size, void* d_ws, size_t ws_size,
                              hipStream_t stream) {
    const float*     x    = (const float*)d_in[0];
    const long long* ei   = (const long long*)d_in[1];   // int64 edge_index [2,E]
    const long long* src  = ei;
    const long long* dst  = ei + NEDGES;
    const float*     W1_l = (const float*)d_in[2];
    const float*     b1   = (const float*)d_in[3];
    const float*     W1_r = (const float*)d_in[4];
    const float*     W2_l = (const float*)d_in[5];
    const float*     b2   = (const float*)d_in[6];
    const float*     W2_r = (const float*)d_in[7];
    float*           out  = (float*)d_out;

    // workspace layout: agg [N*64] | h [N*64] | deg [N]
    float* agg = (float*)d_ws;
    float* h   = agg + (size_t)NNODES * 64;
    float* deg = h   + (size_t)NNODES * 64;

    const int wavesPerBlock = 8; // 256 threads
    const int edgeBlocks = (NEDGES + wavesPerBlock - 1) / wavesPerBlock;
    const int ntiles = NNODES / 16;                       // 6250 exactly
    const int tileBlocks = (ntiles + wavesPerBlock - 1) / wavesPerBlock;

    // Layer 1: mean-aggregate x, then fused WMMA SAGEConv + ReLU
    zero_f32<<<2048, 256, 0, stream>>>(agg, NNODES * 64);
    zero_f32<<<256, 256, 0, stream>>>(deg, NNODES);
    edge_scatter<<<edgeBlocks, 256, 0, stream>>>(x, src, dst, agg, deg, NEDGES);
    sage_layer1_wmma<<<tileBlocks, 256, 0, stream>>>(agg, x, deg, W1_l, b1, W1_r,
                                                     h, ntiles);

    // Layer 2: mean-aggregate h, fused matvec epilogue
    zero_f32<<<2048, 256, 0, stream>>>(agg, NNODES * 64);
    edge_scatter<<<edgeBlocks, 256, 0, stream>>>(h, src, dst, agg, nullptr, NEDGES);
    sage_out<<<(NNODES + 255) / 256, 256, 0, stream>>>(agg, h, deg, W2_l, b2, W2_r,
                                                       out, NNODES);
}